// GraphGaussconv_41188736368645
// MI455X (gfx1250) — compile-verified
//
#include <hip/hip_runtime.h>
#include <math.h>

// Problem constants (fixed by the reference).
#define B_    4
#define N_    32768
#define K_    16
#define CIN_  32
#define H_    32
#define COUT_ 32
#define BN_   (B_ * N_)         // 131072 nodes total
#define E_    (BN_ * K_)        // 2097152 edges

typedef float v2f __attribute__((ext_vector_type(2)));
typedef float v8f __attribute__((ext_vector_type(8)));

__device__ __forceinline__ v8f wmma_f32_k4(v2f a, v2f b, v8f c) {
    // D(16x16,f32) = A(16x4,f32) x B(4x16,f32) + C
    return __builtin_amdgcn_wmma_f32_16x16x4_f32(
        /*neg_a=*/false, a, /*neg_b=*/false, b,
        /*c_mod=*/(short)0, c, /*reuse_a=*/false, /*reuse_b=*/false);
}

// ---------------------------------------------------------------------------
// Kernel 0: zero the accumulation buffer (needed every call; segment-sum
// starts from zero and graph replays do not re-poison the workspace).
// ---------------------------------------------------------------------------
__global__ void gg_zero_kernel(float4* __restrict__ p, int n4) {
    int i = blockIdx.x * blockDim.x + threadIdx.x;
    if (i < n4) p[i] = make_float4(0.f, 0.f, 0.f, 0.f);
}

// ---------------------------------------------------------------------------
// Kernel 1: fc1.  hw[node, h] = (sum_c x[b,c,n]*W1[h,c] + b1[h]) * gw[node]
// One wave handles 16 nodes; 16 v_wmma_f32_16x16x4_f32 per wave.
// A layout (32-bit, 16x4): lanes 0-15 hold K=0,1 ; lanes 16-31 hold K=2,3.
// D layout (32-bit, 16x16): VGPR r, lanes 0-15 -> M=r ; lanes 16-31 -> M=r+8.
// ---------------------------------------------------------------------------
__global__ void gg_fc1_kernel(const float* __restrict__ x,
                              const float* __restrict__ W1,
                              const float* __restrict__ bias1,
                              const float* __restrict__ gw,
                              float* __restrict__ hw) {
    int tid   = blockIdx.x * blockDim.x + threadIdx.x;
    int wave  = tid >> 5;
    int lane  = tid & 31;
    int node0 = wave << 4;                 // first of 16 nodes (global over B*N)
    if (node0 >= BN_) return;

    int m  = lane & 15;                    // row within A / column within B,D
    int kh = lane >> 4;                    // 0 -> K 0,1 ; 1 -> K 2,3
    int b  = node0 / N_;                   // 16-node tile never crosses batch
    int nloc = node0 - b * N_ + m;
    const float* xb = x + (size_t)b * CIN_ * N_;

    v8f acc0 = {0.f, 0.f, 0.f, 0.f, 0.f, 0.f, 0.f, 0.f};
    v8f acc1 = acc0;

#pragma unroll
    for (int k0 = 0; k0 < CIN_; k0 += 4) {
        int c0 = k0 + (kh << 1);
        v2f a;
        a.x = xb[(size_t)(c0 + 0) * N_ + nloc];
        a.y = xb[(size_t)(c0 + 1) * N_ + nloc];
        v2f bA, bB;                        // B[K=c, N=h] = W1[h, c]
        bA.x = W1[(size_t)m * CIN_ + c0];
        bA.y = W1[(size_t)m * CIN_ + c0 + 1];
        bB.x = W1[(size_t)(m + 16) * CIN_ + c0];
        bB.y = W1[(size_t)(m + 16) * CIN_ + c0 + 1];
        acc0 = wmma_f32_k4(a, bA, acc0);   // hidden 0..15
        acc1 = wmma_f32_k4(a, bB, acc1);   // hidden 16..31
    }

    float biasLo = bias1[m];
    float biasHi = bias1[m + 16];
#pragma unroll
    for (int r = 0; r < 8; ++r) {
        int nr = node0 + r + (kh << 3);    // this lane's D rows
        float g = gw[nr];
        hw[(size_t)nr * H_ + m]      = (acc0[r] + biasLo) * g;
        hw[(size_t)nr * H_ + m + 16] = (acc1[r] + biasHi) * g;
    }
}

// ---------------------------------------------------------------------------
// Kernel 2: edge stage. One wave per edge, lane == hidden channel.
// acc[ed, lane] += sqrt((bw/pi)^3) * exp(-bw * d2(es,ed)) * hw[es, lane]
// 32 lanes touch exactly one 128B line on gather and on atomic scatter.
// ---------------------------------------------------------------------------
__global__ void gg_edge_kernel(const float* __restrict__ grid,
                               const int* __restrict__ esrc,
                               const int* __restrict__ edst,
                               const float* __restrict__ baseweight,
                               const float* __restrict__ hw,
                               float* __restrict__ acc) {
    int lane   = threadIdx.x & 31;
    int wave   = (blockIdx.x * blockDim.x + threadIdx.x) >> 5;
    int nwaves = (gridDim.x * blockDim.x) >> 5;

    float bw = baseweight[lane];
    const float INV_PI = 0.31830988618379067154f;
    float t    = bw * INV_PI;
    float coef = sqrtf(t * t * t);         // sqrt((bw/pi)^3)

    for (int e = wave; e < E_; e += nwaves) {
        int b   = e / (N_ * K_);           // power-of-two divide -> shift
        int off = b * N_;
        int es  = esrc[e] + off;
        int ed  = edst[e] + off;
        float dx = grid[es * 3 + 0] - grid[ed * 3 + 0];
        float dy = grid[es * 3 + 1] - grid[ed * 3 + 1];
        float dz = grid[es * 3 + 2] - grid[ed * 3 + 2];
        float d2 = dx * dx + dy * dy + dz * dz;
        float gauss = coef * __expf(-bw * d2);
        float v = gauss * hw[(size_t)es * H_ + lane];
        atomicAdd(&acc[(size_t)ed * H_ + lane], v);
    }
}

// ---------------------------------------------------------------------------
// Kernel 3: fc2.  out[b, o, n] = sum_h acc[node,h]*W2[o,h] + b2[o]
// Same WMMA tiling as fc1; A rows now contiguous (stride 32) -> v2f loads.
// ---------------------------------------------------------------------------
__global__ void gg_fc2_kernel(const float* __restrict__ acc,
                              const float* __restrict__ W2,
                              const float* __restrict__ bias2,
                              float* __restrict__ out) {
    int tid   = blockIdx.x * blockDim.x + threadIdx.x;
    int wave  = tid >> 5;
    int lane  = tid & 31;
    int node0 = wave << 4;
    if (node0 >= BN_) return;

    int m  = lane & 15;
    int kh = lane >> 4;
    const float* arow = acc + (size_t)(node0 + m) * H_;

    v8f d0 = {0.f, 0.f, 0.f, 0.f, 0.f, 0.f, 0.f, 0.f};
    v8f d1 = d0;

#pragma unroll
    for (int k0 = 0; k0 < H_; k0 += 4) {
        int c0 = k0 + (kh << 1);           // even -> 8B-aligned v2f load
        v2f a = *(const v2f*)(arow + c0);
        v2f bA, bB;                        // B[K=h, N=o] = W2[o, h]
        bA.x = W2[(size_t)m * H_ + c0];
        bA.y = W2[(size_t)m * H_ + c0 + 1];
        bB.x = W2[(size_t)(m + 16) * H_ + c0];
        bB.y = W2[(size_t)(m + 16) * H_ + c0 + 1];
        d0 = wmma_f32_k4(a, bA, d0);       // cout 0..15
        d1 = wmma_f32_k4(a, bB, d1);       // cout 16..31
    }

    int b    = node0 / N_;
    int nl0  = node0 - b * N_;
    float bLo = bias2[m];
    float bHi = bias2[m + 16];
    float* ob = out + (size_t)b * COUT_ * N_;
#pragma unroll
    for (int r = 0; r < 8; ++r) {
        int nl = nl0 + r + (kh << 3);
        ob[(size_t)m * N_ + nl]        = d0[r] + bLo;   // out[b, m, nl]
        ob[(size_t)(m + 16) * N_ + nl] = d1[r] + bHi;   // out[b, m+16, nl]
    }
}

// ---------------------------------------------------------------------------
// Launch
// ---------------------------------------------------------------------------
extern "C" void kernel_launch(void* const* d_in, const int* in_sizes, int n_in,
                              void* d_out, int out_size, void* d_ws, size_t ws_size,
                              hipStream_t stream) {
    const float* x    = (const float*)d_in[0];   // (B, CIN, N)
    const float* grid = (const float*)d_in[1];   // (B, N, P)
    const float* gw   = (const float*)d_in[2];   // (B, N)
    const int*   esrc = (const int*)d_in[3];     // (B, N, K)
    const int*   edst = (const int*)d_in[4];     // (B, N, K)
    const float* W1   = (const float*)d_in[5];   // (H, CIN)
    const float* b1   = (const float*)d_in[6];   // (H,)
    const float* W2   = (const float*)d_in[7];   // (COUT, H)
    const float* b2   = (const float*)d_in[8];   // (COUT,)
    const float* bwv  = (const float*)d_in[9];   // (H,) baseweight
    float* out = (float*)d_out;                  // (B, COUT, N)

    float* hw   = (float*)d_ws;                  // (B*N, H)  = 16 MB
    float* accb = hw + (size_t)BN_ * H_;         // (B*N, H)  = 16 MB

    // 0) zero segment-sum accumulator
    int n4 = (BN_ * H_) / 4;
    gg_zero_kernel<<<(n4 + 255) / 256, 256, 0, stream>>>((float4*)accb, n4);

    // 1) fc1 (+bias, * source grid_weight) -> hw
    // BN_/16 = 8192 waves; 8 waves per 256-thread block -> 1024 blocks
    gg_fc1_kernel<<<1024, 256, 0, stream>>>(x, W1, b1, gw, hw);

    // 2) edge gather / gaussian / scatter-atomic-add
    // 8192 blocks * 8 waves = 65536 waves, grid-stride over 2M edges
    gg_edge_kernel<<<8192, 256, 0, stream>>>(grid, esrc, edst, bwv, hw, accb);

    // 3) fc2 (+bias) -> out (B, COUT, N)
    gg_fc2_kernel<<<1024, 256, 0, stream>>>(accb, W2, b2, out);
}